// MyGCN_4406636445725
// MI455X (gfx1250) — compile-verified
//
#include <hip/hip_runtime.h>
#include <hip/hip_bf16.h>

typedef __attribute__((ext_vector_type(2))) float v2f;
typedef __attribute__((ext_vector_type(8))) float v8f;

#define F_IN 512
#define HID  16

// Non-returning fp32 scatter-add, guaranteed native L2 atomic on gfx1250.
// (GLOBAL_ATOMIC_ADD_F32, TH[0]=0 -> no data return, STOREcnt-tracked.)
__device__ __forceinline__ void atomAddF(float* p, float v) {
    asm volatile("global_atomic_add_f32 %0, %1, off"
                 :
                 : "v"(p), "v"(v)
                 : "memory");
}

// ---------------------------------------------------------------------------
// Degree count: deg[dst]++ for every real edge (self-loop folded in later).
// ---------------------------------------------------------------------------
__global__ void gcn_degree_kernel(unsigned* __restrict__ cnt,
                                  const int* __restrict__ dst, int E) {
    int e = blockIdx.x * blockDim.x + threadIdx.x;
    if (e >= E) return;
    atomicAdd(&cnt[dst[e]], 1u);
}

// dis[i] = rsqrt(deg[i] + 1)   (the +1 is the self-loop; deg >= 1 always)
__global__ void gcn_dis_kernel(float* __restrict__ buf, int n) {
    int i = blockIdx.x * blockDim.x + threadIdx.x;
    if (i >= n) return;
    unsigned c = ((const unsigned*)buf)[i];
    buf[i] = __frsqrt_rn((float)(c + 1u));
}

// ---------------------------------------------------------------------------
// Layer 1 GEMM via WMMA f32 16x16x4:  s1 = dis ⊙ (x @ W1)
// One wave per 16-row tile; W1 (512x16 f32 = 32KB) staged in LDS.
// ---------------------------------------------------------------------------
__global__ __launch_bounds__(256) void gcn_gemm1_kernel(
        const float* __restrict__ x, const float* __restrict__ W1,
        const float* __restrict__ dis, float* __restrict__ s1, int nTiles) {
    __shared__ float wlds[F_IN * HID];
    int t = threadIdx.x;
    // cooperative 128-bit loads of W1 into LDS
    const float4* wg = (const float4*)W1;
    float4* ws4 = (float4*)wlds;
    #pragma unroll 2
    for (int i = t; i < (F_IN * HID) / 4; i += 256) ws4[i] = wg[i];
    __syncthreads();

    int wave = t >> 5;
    int lane = t & 31;
    int lrow = lane & 15;   // A row / B-C column
    int half = lane >> 4;   // K-half selector
    int tile = blockIdx.x * 8 + wave;
    if (tile >= nTiles) return;           // after the only barrier: safe
    int m0 = tile * 16;

    const float* xrow = x + (size_t)(m0 + lrow) * F_IN + 2 * half;
    v8f acc = {};
    #pragma unroll 4
    for (int k0 = 0; k0 < F_IN; k0 += 4) {
        v2f a;                                  // A 16x4: lanes<16 K=k0,k0+1 ; lanes>=16 K=k0+2,k0+3
        a.x = xrow[k0];
        a.y = xrow[k0 + 1];
        int kb = k0 + 2 * half;                 // B 4x16: N=lrow, K=kb,kb+1
        v2f b;
        b.x = wlds[kb * HID + lrow];
        b.y = wlds[(kb + 1) * HID + lrow];
        acc = __builtin_amdgcn_wmma_f32_16x16x4_f32(
            false, a, false, b, (short)0, acc, false, false);
    }
    // C layout: VGPR r -> row m0 + r + 8*half, col lrow. Pre-scale by dis[row].
    #pragma unroll
    for (int r = 0; r < 8; ++r) {
        int row = m0 + r + 8 * half;
        s1[(size_t)row * HID + lrow] = dis[row] * acc[r];
    }
}

// ---------------------------------------------------------------------------
// Edge scatter:  agg[dst] += s[src]   (16 features per edge, fp32 atomics in L2)
// ---------------------------------------------------------------------------
__global__ void gcn_scatter_kernel(const float* __restrict__ sv,
                                   const int* __restrict__ src,
                                   const int* __restrict__ dst,
                                   float* __restrict__ agg, int E) {
    int e = blockIdx.x * blockDim.x + threadIdx.x;
    if (e >= E) return;
    int s = src[e];
    int d = dst[e];
    const float4* sp = (const float4*)(sv + (size_t)s * HID);
    float* dp = agg + (size_t)d * HID;
    #pragma unroll
    for (int q = 0; q < 4; ++q) {
        float4 v = sp[q];
        atomAddF(dp + 4 * q + 0, v.x);
        atomAddF(dp + 4 * q + 1, v.y);
        atomAddF(dp + 4 * q + 2, v.z);
        atomAddF(dp + 4 * q + 3, v.w);
    }
}

// ---------------------------------------------------------------------------
// Layer-1 epilogue + Layer-2 GEMM (WMMA) + pre-scale:
//   h   = elu(dis*(agg1 + s1) + b1)          (self-loop folded: + s1 term)
//   s2  = dis ⊙ (h @ W2)
// One wave per 16-node tile; h tile transposed through LDS into A layout.
// ---------------------------------------------------------------------------
__global__ __launch_bounds__(256) void gcn_layer2_kernel(
        const float* __restrict__ agg1, const float* __restrict__ s1,
        const float* __restrict__ dis, const float* __restrict__ b1,
        const float* __restrict__ W2, float* __restrict__ s2, int nTiles) {
    __shared__ float w2s[HID * HID];      // 1 KB
    __shared__ float hbuf[8 * 16 * HID];  // 8 KB, one 16x16 tile per wave
    int t = threadIdx.x;
    w2s[t] = W2[t];                       // 256 threads == 256 elements

    int wave = t >> 5;
    int lane = t & 31;
    int lrow = lane & 15;
    int half = lane >> 4;
    int tile = blockIdx.x * 8 + wave;
    if (tile >= nTiles) tile = nTiles - 1;   // keep all waves at the barrier
    int m0 = tile * 16;

    float* hb = &hbuf[wave * 16 * HID];
    float bv = b1[lrow];
    // produce h in C-layout positions, write row-major into LDS for transpose
    #pragma unroll
    for (int r = 0; r < 8; ++r) {
        int row = m0 + r + 8 * half;
        float dv = dis[row];
        size_t o = (size_t)row * HID + lrow;
        float z = dv * (agg1[o] + s1[o]) + bv;
        float h = z > 0.0f ? z : expm1f(z);
        hb[(r + 8 * half) * HID + lrow] = h;
    }
    __syncthreads();

    v8f acc = {};
    #pragma unroll
    for (int k0 = 0; k0 < HID; k0 += 4) {
        int ka = k0 + 2 * half;
        v2f a;
        a.x = hb[lrow * HID + ka];
        a.y = hb[lrow * HID + ka + 1];
        v2f b;
        b.x = w2s[ka * HID + lrow];
        b.y = w2s[(ka + 1) * HID + lrow];
        acc = __builtin_amdgcn_wmma_f32_16x16x4_f32(
            false, a, false, b, (short)0, acc, false, false);
    }
    #pragma unroll
    for (int r = 0; r < 8; ++r) {
        int row = m0 + r + 8 * half;
        s2[(size_t)row * HID + lrow] = dis[row] * acc[r];
    }
}

// ---------------------------------------------------------------------------
// Final: out = log_softmax(dis*(agg2 + s2) + b2), one thread per node.
// ---------------------------------------------------------------------------
__global__ void gcn_final_kernel(const float* __restrict__ agg2,
                                 const float* __restrict__ s2,
                                 const float* __restrict__ dis,
                                 const float* __restrict__ b2,
                                 float* __restrict__ out, int n) {
    int i = blockIdx.x * blockDim.x + threadIdx.x;
    if (i >= n) return;
    float dv = dis[i];
    float z[HID];
    float m = -3.402823e38f;
    #pragma unroll
    for (int f = 0; f < HID; ++f) {
        size_t o = (size_t)i * HID + f;
        float v = dv * (agg2[o] + s2[o]) + b2[f];
        z[f] = v;
        m = fmaxf(m, v);
    }
    float ssum = 0.0f;
    #pragma unroll
    for (int f = 0; f < HID; ++f) ssum += __expf(z[f] - m);
    float lse = m + __logf(ssum);
    #pragma unroll
    for (int f = 0; f < HID; ++f) out[(size_t)i * HID + f] = z[f] - lse;
}

// ---------------------------------------------------------------------------
extern "C" void kernel_launch(void* const* d_in, const int* in_sizes, int n_in,
                              void* d_out, int out_size, void* d_ws, size_t ws_size,
                              hipStream_t stream) {
    const float* x  = (const float*)d_in[0];
    const int*   ei = (const int*)d_in[1];
    const float* W1 = (const float*)d_in[2];
    const float* b1 = (const float*)d_in[3];
    const float* W2 = (const float*)d_in[4];
    const float* b2 = (const float*)d_in[5];
    float* out = (float*)d_out;

    int n = in_sizes[0] / F_IN;       // 100000 nodes
    int E = in_sizes[1] / 2;          // 3200000 edges
    const int* src = ei;
    const int* dst = ei + E;

    float* ws   = (float*)d_ws;
    float* dis  = ws;                           // [n]      (starts life as uint counts)
    float* s1   = dis  + n;                     // [n*16]
    float* agg1 = s1   + (size_t)n * HID;       // [n*16]
    float* s2   = agg1 + (size_t)n * HID;       // [n*16]
    float* agg2 = s2   + (size_t)n * HID;       // [n*16]

    hipMemsetAsync(dis,  0, (size_t)n * sizeof(float), stream);
    hipMemsetAsync(agg1, 0, (size_t)n * HID * sizeof(float), stream);
    hipMemsetAsync(agg2, 0, (size_t)n * HID * sizeof(float), stream);

    int tB = 256;
    int nTiles = (n + 15) / 16;

    gcn_degree_kernel<<<(E + tB - 1) / tB, tB, 0, stream>>>((unsigned*)dis, dst, E);
    gcn_dis_kernel<<<(n + tB - 1) / tB, tB, 0, stream>>>(dis, n);
    gcn_gemm1_kernel<<<(nTiles + 7) / 8, tB, 0, stream>>>(x, W1, dis, s1, nTiles);
    gcn_scatter_kernel<<<(E + tB - 1) / tB, tB, 0, stream>>>(s1, src, dst, agg1, E);
    gcn_layer2_kernel<<<(nTiles + 7) / 8, tB, 0, stream>>>(agg1, s1, dis, b1, W2, s2, nTiles);
    gcn_scatter_kernel<<<(E + tB - 1) / tB, tB, 0, stream>>>(s2, src, dst, agg2, E);
    gcn_final_kernel<<<(n + tB - 1) / tB, tB, 0, stream>>>(agg2, s2, dis, b2, out, n);
}